// TGNN_34875134443608
// MI455X (gfx1250) — compile-verified
//
#include <hip/hip_runtime.h>
#include <hip/hip_bf16.h>
#include <math.h>

// ---------------- problem constants (match reference) ----------------
#define NN    2048
#define TT    1024
#define EE    16384
#define BB    32
#define C1    8
#define C2    16
#define C3    32
#define K1    25
#define K2    15
#define K3    7
#define T2    61
#define DD    128
#define FEAT  (DD * T2)            // 7808
#define MROWS (NN * T2)            // 124928 rows for GCN GEMMs (divisible by 16)
#define EPSV  1e-5f

typedef float v2f __attribute__((ext_vector_type(2)));
typedef float v8f __attribute__((ext_vector_type(8)));

// ---------------- utility ----------------
__global__ void zero_kernel(float* __restrict__ p, long n) {
    long i = (long)blockIdx.x * blockDim.x + threadIdx.x;
    long stride = (long)gridDim.x * blockDim.x;
    for (; i < n; i += stride) p[i] = 0.0f;
}

// ---------------- conv1 (+BN fold +ReLU), pad K1/2, out: [n][t][c1] ----------------
__global__ __launch_bounds__(256)
void conv1_kernel(const float* __restrict__ x, const float* __restrict__ w,
                  const float* __restrict__ cb, const float* __restrict__ g,
                  const float* __restrict__ bb, float* __restrict__ h1) {
    __shared__ float wf[C1 * K1];
    __shared__ float bf[C1];
    const float rs = rsqrtf(1.0f + EPSV);
    int tid = threadIdx.x;
    if (tid < C1 * K1) { int c = tid / K1; wf[tid] = w[tid] * g[c] * rs; }
    if (tid < C1)      { bf[tid] = cb[tid] * g[tid] * rs + bb[tid]; }
    __syncthreads();
    long idx = (long)blockIdx.x * 256 + tid;
    if (idx >= (long)NN * TT) return;
    int n = (int)(idx / TT), t = (int)(idx % TT);
    float acc[C1];
#pragma unroll
    for (int c = 0; c < C1; ++c) acc[c] = bf[c];
    const float* xp = x + (long)n * TT;
#pragma unroll
    for (int k = 0; k < K1; ++k) {
        int tt = t + k - (K1 / 2);
        if (tt >= 0 && tt < TT) {
            float xv = xp[tt];
#pragma unroll
            for (int c = 0; c < C1; ++c) acc[c] = fmaf(xv, wf[c * K1 + k], acc[c]);
        }
    }
    float* op = h1 + idx * C1;
#pragma unroll
    for (int c = 0; c < C1; ++c) op[c] = fmaxf(acc[c], 0.0f);
}

// ---------------- positional MLP: pe = relu(pos@W1+b1)@W2+b2 ----------------
__global__ __launch_bounds__(256)
void pe_kernel(const float* __restrict__ pos, const float* __restrict__ w1,
               const float* __restrict__ b1, const float* __restrict__ w2,
               const float* __restrict__ b2, float* __restrict__ pe) {
    int n = blockIdx.x * blockDim.x + threadIdx.x;
    if (n >= NN) return;
    float p0 = pos[3 * n + 0], p1 = pos[3 * n + 1], p2 = pos[3 * n + 2];
    float hid[C3];
#pragma unroll
    for (int j = 0; j < C3; ++j) {
        float v = p0 * w1[j] + p1 * w1[C3 + j] + p2 * w1[2 * C3 + j] + b1[j];
        hid[j] = fmaxf(v, 0.0f);
    }
    float* op = pe + (long)n * C3;
    for (int j = 0; j < C3; ++j) {
        float a = b2[j];
#pragma unroll
        for (int i = 0; i < C3; ++i) a = fmaf(hid[i], w2[i * C3 + j], a);
        op[j] = a;
    }
}

// ---------------- conv2 (+BN fold +avgpool4 +ReLU), out: [n][tp][c2], tp<252 ----------------
__global__ __launch_bounds__(256)
void conv2_kernel(const float* __restrict__ h1, const float* __restrict__ w,
                  const float* __restrict__ cb, const float* __restrict__ g,
                  const float* __restrict__ bb, float* __restrict__ h2) {
    __shared__ float wf[C2 * C1 * K2];   // 1920
    __shared__ float bf[C2];
    const float rs = rsqrtf(1.0f + EPSV);
    int tid = threadIdx.x;
    for (int i = tid; i < C2 * C1 * K2; i += 256) {
        int c = i / (C1 * K2);
        wf[i] = w[i] * g[c] * rs * 0.25f;      // fold BN scale and 1/4 pooling
    }
    if (tid < C2) bf[tid] = cb[tid] * g[tid] * rs + bb[tid];
    __syncthreads();
    long idx = (long)blockIdx.x * 256 + tid;
    if (idx >= (long)NN * 252) return;
    int n = (int)(idx / 252), tp = (int)(idx % 252);
    float acc[C2];
#pragma unroll
    for (int c = 0; c < C2; ++c) acc[c] = bf[c];
    const float* base = h1 + ((long)n * TT + tp * 4) * C1;
#pragma unroll
    for (int p = 0; p < 4; ++p) {
        for (int k = 0; k < K2; ++k) {
            const float* hp = base + (long)(p + k) * C1;
#pragma unroll
            for (int ci = 0; ci < C1; ++ci) {
                float v = hp[ci];
#pragma unroll
                for (int c = 0; c < C2; ++c)
                    acc[c] = fmaf(v, wf[(c * C1 + ci) * K2 + k], acc[c]);
            }
        }
    }
    float* op = h2 + idx * C2;
#pragma unroll
    for (int c = 0; c < C2; ++c) op[c] = fmaxf(acc[c], 0.0f);
}

// ---------------- conv3 (+BN fold +avgpool4 +ReLU +pe add), out X: [n*61+t2][c3] ----------------
__global__ __launch_bounds__(256)
void conv3_kernel(const float* __restrict__ h2, const float* __restrict__ w,
                  const float* __restrict__ cb, const float* __restrict__ g,
                  const float* __restrict__ bb, const float* __restrict__ pe,
                  float* __restrict__ X) {
    __shared__ float wf[C3 * C2 * K3];   // 3584
    __shared__ float bf[C3];
    const float rs = rsqrtf(1.0f + EPSV);
    int tid = threadIdx.x;
    for (int i = tid; i < C3 * C2 * K3; i += 256) {
        int c = i / (C2 * K3);
        wf[i] = w[i] * g[c] * rs * 0.25f;
    }
    if (tid < C3) bf[tid] = cb[tid] * g[tid] * rs + bb[tid];
    __syncthreads();
    long idx = (long)blockIdx.x * 256 + tid;
    if (idx >= (long)NN * T2) return;
    int n = (int)(idx / T2), t2 = (int)(idx % T2);
    float acc[C3];
#pragma unroll
    for (int c = 0; c < C3; ++c) acc[c] = bf[c];
    const float* base = h2 + ((long)n * 252 + t2 * 4) * C2;
#pragma unroll
    for (int p = 0; p < 4; ++p) {
#pragma unroll
        for (int k = 0; k < K3; ++k) {
            const float* hp = base + (long)(p + k) * C2;
#pragma unroll
            for (int ci = 0; ci < C2; ++ci) {
                float v = hp[ci];
#pragma unroll
                for (int c = 0; c < C3; ++c)
                    acc[c] = fmaf(v, wf[(c * C2 + ci) * K3 + k], acc[c]);
            }
        }
    }
    const float* pep = pe + (long)n * C3;
    float* op = X + idx * C3;
#pragma unroll
    for (int c = 0; c < C3; ++c) op[c] = fmaxf(acc[c], 0.0f) + pep[c];
}

// ---------------- degree / dinv ----------------
__global__ void deg_kernel(const int* __restrict__ ei, float* __restrict__ deg) {
    int e = blockIdx.x * blockDim.x + threadIdx.x;
    if (e >= EE) return;
    unsafeAtomicAdd(&deg[ei[EE + e]], 1.0f);   // dst in-degree
}
__global__ void dinv_kernel(const float* __restrict__ deg, float* __restrict__ dinv) {
    int n = blockIdx.x * blockDim.x + threadIdx.x;
    if (n >= NN) return;
    dinv[n] = rsqrtf(deg[n] + 1.0f);           // +1 self loop
}

// ---------------- GCN GEMM via V_WMMA_F32_16X16X4_F32 ----------------
// X: [MROWS x K] row-major, W: [K x 128] row-major.
// out[row][col] = (X@W)[row][col] * dinv[row/61]   (left dinv scaling fused)
// One wave per 16x16 tile; 8 waves/block cover one full 128-col tile-row.
template <int K>
__global__ __launch_bounds__(256)
void gemm_wmma_kernel(const float* __restrict__ X, const float* __restrict__ W,
                      const float* __restrict__ dinv, float* __restrict__ out) {
    const int lane = threadIdx.x & 31;
    const int wv   = threadIdx.x >> 5;        // column tile 0..7
    const int r    = lane & 15;
    const int hs   = lane >> 4;               // half-select (0: K,K+1 | 1: K+2,K+3)
    const long mbase = (long)blockIdx.x * 16;
    const float* xrow = X + (mbase + r) * K;
    const int col = wv * 16 + r;
    v8f acc = {};
#pragma unroll
    for (int kk = 0; kk < K; kk += 4) {
        const int ka = kk + hs * 2;
        v2f a = { xrow[ka], xrow[ka + 1] };
        v2f b = { W[(long)ka * DD + col], W[(long)(ka + 1) * DD + col] };
        acc = __builtin_amdgcn_wmma_f32_16x16x4_f32(
            /*neg_a=*/false, a, /*neg_b=*/false, b,
            /*c_mod=*/(short)0, acc, /*reuse_a=*/false, /*reuse_b=*/false);
    }
#pragma unroll
    for (int j = 0; j < 8; ++j) {
        long row = mbase + j + hs * 8;
        out[row * DD + col] = acc[j] * dinv[row / T2];
    }
}

// ---------------- edge scatter: agg[dst] += hw[src] (per-node 61x128 block) ----------------
__global__ __launch_bounds__(256)
void edge_scatter_kernel(const int* __restrict__ ei, const float* __restrict__ hw,
                         float* __restrict__ agg) {
    int e = blockIdx.x;
    int s = ei[e], d = ei[EE + e];
    const float* sp = hw + (long)s * FEAT;
    float* dp = agg + (long)d * FEAT;
    for (int f = threadIdx.x; f < FEAT; f += 256)
        unsafeAtomicAdd(&dp[f], sp[f]);
}

// ---------------- GCN epilogue: relu((agg + hw_selfloop) * dinv + b) ----------------
__global__ __launch_bounds__(256)
void gcn_epilogue_kernel(const float* __restrict__ agg, const float* __restrict__ hw,
                         const float* __restrict__ dinv, const float* __restrict__ b,
                         float* __restrict__ out) {
    long idx = (long)blockIdx.x * 256 + threadIdx.x;
    if (idx >= (long)MROWS * DD) return;
    long row = idx >> 7;
    int c = (int)(idx & (DD - 1));
    float v = (agg[idx] + hw[idx]) * dinv[row / T2] + b[c];
    out[idx] = fmaxf(v, 0.0f);
}

// ---------------- batch pooling ----------------
__global__ void pool_cnt_kernel(const int* __restrict__ batch, float* __restrict__ cnt) {
    int n = blockIdx.x * blockDim.x + threadIdx.x;
    if (n >= NN) return;
    unsafeAtomicAdd(&cnt[batch[n]], 1.0f);
}
__global__ __launch_bounds__(256)
void pool_feat_kernel(const float* __restrict__ h, const int* __restrict__ batch,
                      float* __restrict__ pooled) {
    long idx = (long)blockIdx.x * 256 + threadIdx.x;
    if (idx >= (long)NN * FEAT) return;
    int n = (int)(idx / FEAT);
    int rem = (int)(idx % FEAT);
    int t = rem >> 7, c = rem & (DD - 1);
    // feat[n][c*61+t] = h[(n*61+t)*128 + c]  (== h[idx])
    unsafeAtomicAdd(&pooled[(long)batch[n] * FEAT + c * T2 + t], h[idx]);
}

// ---------------- classifier + log_softmax ----------------
__global__ __launch_bounds__(128)
void final_kernel(const float* __restrict__ pooled, const float* __restrict__ cnt,
                  const float* __restrict__ dw, const float* __restrict__ db,
                  float* __restrict__ out) {
    __shared__ float red[128][4];
    int b = blockIdx.x, tid = threadIdx.x;
    float a[4] = {0.f, 0.f, 0.f, 0.f};
    const float* pp = pooled + (long)b * FEAT;
    for (int k = tid; k < FEAT; k += 128) {
        float pv = pp[k];
#pragma unroll
        for (int j = 0; j < 4; ++j) a[j] = fmaf(pv, dw[k * 4 + j], a[j]);
    }
#pragma unroll
    for (int j = 0; j < 4; ++j) red[tid][j] = a[j];
    __syncthreads();
    for (int off = 64; off > 0; off >>= 1) {
        if (tid < off)
#pragma unroll
            for (int j = 0; j < 4; ++j) red[tid][j] += red[tid + off][j];
        __syncthreads();
    }
    if (tid == 0) {
        float inv = 1.0f / fmaxf(cnt[b], 1.0f);
        float l[4], m = -1e30f;
#pragma unroll
        for (int j = 0; j < 4; ++j) { l[j] = red[0][j] * inv + db[j]; m = fmaxf(m, l[j]); }
        float s = 0.f;
#pragma unroll
        for (int j = 0; j < 4; ++j) s += __expf(l[j] - m);
        float lse = m + logf(s);
#pragma unroll
        for (int j = 0; j < 4; ++j) out[b * 4 + j] = l[j] - lse;
    }
}

// ---------------- host orchestration ----------------
extern "C" void kernel_launch(void* const* d_in, const int* in_sizes, int n_in,
                              void* d_out, int out_size, void* d_ws, size_t ws_size,
                              hipStream_t stream) {
    const float* x       = (const float*)d_in[0];
    const float* pos     = (const float*)d_in[1];
    const float* conv1_w = (const float*)d_in[2];
    const float* conv1_b = (const float*)d_in[3];
    const float* bn1_g   = (const float*)d_in[4];
    const float* bn1_b   = (const float*)d_in[5];
    const float* conv2_w = (const float*)d_in[6];
    const float* conv2_b = (const float*)d_in[7];
    const float* bn2_g   = (const float*)d_in[8];
    const float* bn2_b   = (const float*)d_in[9];
    const float* conv3_w = (const float*)d_in[10];
    const float* conv3_b = (const float*)d_in[11];
    const float* bn3_g   = (const float*)d_in[12];
    const float* bn3_b   = (const float*)d_in[13];
    const float* pos_w1  = (const float*)d_in[14];
    const float* pos_b1  = (const float*)d_in[15];
    const float* pos_w2  = (const float*)d_in[16];
    const float* pos_b2  = (const float*)d_in[17];
    const float* gcn1_w  = (const float*)d_in[18];
    const float* gcn1_b  = (const float*)d_in[19];
    const float* gcn2_w  = (const float*)d_in[20];
    const float* gcn2_b  = (const float*)d_in[21];
    const float* dense_w = (const float*)d_in[22];
    const float* dense_b = (const float*)d_in[23];
    const int*   ei      = (const int*)d_in[24];
    const int*   batch   = (const int*)d_in[25];
    float* out = (float*)d_out;

    // workspace slabs (floats); slab A reused h1 -> hw1 -> hw2
    float* ws = (float*)d_ws;
    const long szA  = (long)NN * TT * C1;          // 16,777,216 (>= MROWS*DD)
    const long szB  = (long)MROWS * DD;            // 15,990,784
    const long szC  = (long)MROWS * DD;            // h2 (8.26M) then h_gcn
    const long szD  = (long)MROWS * C3;            // 3,997,696
    float* bufA   = ws;                            // h1 / hw
    float* bufB   = bufA + szA;                    // agg
    float* bufC   = bufB + szB;                    // h2 / h_gcn
    float* bufD   = bufC + szC;                    // X1
    float* pe     = bufD + szD;                    // NN*C3
    float* deg    = pe + (long)NN * C3;            // NN
    float* dinv   = deg + NN;                      // NN
    float* pooled = dinv + NN;                     // BB*FEAT
    float* cnt    = pooled + (long)BB * FEAT;      // BB
    (void)ws_size; (void)in_sizes; (void)n_in; (void)out_size;

    const int ZB = 2048;
    // zero accumulators (required each call: atomics + poisoned workspace)
    zero_kernel<<<ZB, 256, 0, stream>>>(deg, (long)NN);
    zero_kernel<<<ZB, 256, 0, stream>>>(pooled, (long)BB * FEAT);
    zero_kernel<<<1, 32, 0, stream>>>(cnt, (long)BB);

    // stage 1-3: temporal conv stack (BN folded)
    long p1 = (long)NN * TT;
    conv1_kernel<<<(int)((p1 + 255) / 256), 256, 0, stream>>>(x, conv1_w, conv1_b, bn1_g, bn1_b, bufA);
    pe_kernel<<<(NN + 255) / 256, 256, 0, stream>>>(pos, pos_w1, pos_b1, pos_w2, pos_b2, pe);
    long p2 = (long)NN * 252;
    conv2_kernel<<<(int)((p2 + 255) / 256), 256, 0, stream>>>(bufA, conv2_w, conv2_b, bn2_g, bn2_b, bufC);
    long p3 = (long)NN * T2;
    conv3_kernel<<<(int)((p3 + 255) / 256), 256, 0, stream>>>(bufC, conv3_w, conv3_b, bn3_g, bn3_b, pe, bufD);

    // degrees
    deg_kernel<<<(EE + 255) / 256, 256, 0, stream>>>(ei, deg);
    dinv_kernel<<<(NN + 255) / 256, 256, 0, stream>>>(deg, dinv);

    const int MT = MROWS / 16;       // 7808 tile-rows
    const long elems = (long)MROWS * DD;
    const int EPB = (int)((elems + 255) / 256);

    // GCN layer 1: WMMA GEMM (K=32) -> scatter -> epilogue
    gemm_wmma_kernel<C3><<<MT, 256, 0, stream>>>(bufD, gcn1_w, dinv, bufA);
    zero_kernel<<<ZB, 256, 0, stream>>>(bufB, szB);
    edge_scatter_kernel<<<EE, 256, 0, stream>>>(ei, bufA, bufB);
    gcn_epilogue_kernel<<<EPB, 256, 0, stream>>>(bufB, bufA, dinv, gcn1_b, bufC);

    // GCN layer 2: WMMA GEMM (K=128) -> scatter -> epilogue
    gemm_wmma_kernel<DD><<<MT, 256, 0, stream>>>(bufC, gcn2_w, dinv, bufA);
    zero_kernel<<<ZB, 256, 0, stream>>>(bufB, szB);
    edge_scatter_kernel<<<EE, 256, 0, stream>>>(ei, bufA, bufB);
    gcn_epilogue_kernel<<<EPB, 256, 0, stream>>>(bufB, bufA, dinv, gcn2_b, bufC);

    // pooling + classifier
    pool_cnt_kernel<<<(NN + 255) / 256, 256, 0, stream>>>(batch, cnt);
    pool_feat_kernel<<<EPB, 256, 0, stream>>>(bufC, batch, pooled);
    final_kernel<<<BB, 128, 0, stream>>>(pooled, cnt, dense_w, dense_b, out);
}